// GraphEncoder_66537633349984
// MI455X (gfx1250) — compile-verified
//
#include <hip/hip_runtime.h>
#include <hip/hip_bf16.h>

#define DIM 128
#define BN_EPS 1e-5f

typedef __attribute__((ext_vector_type(2))) float v2f;
typedef __attribute__((ext_vector_type(8))) float v8f;

// ---------------------------------------------------------------------------
// 1) init: deg[i] = 1.0 (self loop), agg[j] = 0
// ---------------------------------------------------------------------------
__global__ void init_deg_agg(float* __restrict__ deg, float* __restrict__ agg, int n_nodes) {
    int j = blockIdx.x * blockDim.x + threadIdx.x;
    int total = n_nodes * DIM;
    if (j < total) agg[j] = 0.0f;
    if (j < n_nodes) deg[j] = 1.0f;   // self-loop contributes 1 to degree
}

// ---------------------------------------------------------------------------
// 2) degree accumulation over real edges (non-returning f32 atomics)
// ---------------------------------------------------------------------------
__global__ void accum_deg(const int* __restrict__ dst, float* __restrict__ deg, int n_edges) {
    int e = blockIdx.x * blockDim.x + threadIdx.x;
    if (e < n_edges) unsafeAtomicAdd(&deg[dst[e]], 1.0f);
}

// ---------------------------------------------------------------------------
// 3) dinv[i] = rsqrt(max(deg,1e-12))  (in place)
// ---------------------------------------------------------------------------
__global__ void finish_dinv(float* __restrict__ deg, int n_nodes) {
    int i = blockIdx.x * blockDim.x + threadIdx.x;
    if (i < n_nodes) deg[i] = rsqrtf(fmaxf(deg[i], 1e-12f));
}

// ---------------------------------------------------------------------------
// 4) GEMM: HW = H @ W   (H: n x 128, W: 128 x 128)  full-fp32 WMMA
//    One wave per 16x16 output tile; 8 waves (256 thr) cover one 16-row strip.
//    A-frag layout (ISA 7.12.2, 32-bit A 16x4): lanes 0-15 -> K=k,k+1 ; 16-31 -> K=k+2,k+3
//    B-frag layout (rows striped across lanes): lanes 0-15 -> rows k,k+1 ; 16-31 -> k+2,k+3
//    D layout: VGPR r -> M = r (lanes 0-15) / r+8 (lanes 16-31), N = lane%16
// ---------------------------------------------------------------------------
__global__ void gemm_wmma_f32(const float* __restrict__ H,
                              const float* __restrict__ W,
                              float* __restrict__ HW, int n_nodes) {
    const int row0 = blockIdx.x * 16;           // n_nodes is a multiple of 16 (100000)
    const int wave = threadIdx.x >> 5;          // 0..7 -> 16-col tile
    const int lane = threadIdx.x & 31;
    const int hi   = lane >> 4;                 // 0: lanes 0-15, 1: lanes 16-31
    const int l16  = lane & 15;
    const int col0 = wave * 16;

    const float* arow = H + (size_t)(row0 + l16) * DIM;   // this lane's M row
    v8f c = {0.f, 0.f, 0.f, 0.f, 0.f, 0.f, 0.f, 0.f};

    #pragma unroll
    for (int k = 0; k < DIM; k += 4) {
        const int kk = k + hi * 2;
        v2f a, b;
        a.x = arow[kk + 0];
        a.y = arow[kk + 1];
        b.x = W[(size_t)(kk + 0) * DIM + col0 + l16];
        b.y = W[(size_t)(kk + 1) * DIM + col0 + l16];
        c = __builtin_amdgcn_wmma_f32_16x16x4_f32(
                /*neg_a=*/false, a, /*neg_b=*/false, b,
                /*c_mod=*/(short)0, c, /*reuse_a=*/false, /*reuse_b=*/false);
    }

    float* out = HW + (size_t)row0 * DIM + col0;
    const int m0 = hi * 8;
    #pragma unroll
    for (int r = 0; r < 8; ++r)
        out[(size_t)(m0 + r) * DIM + l16] = c[r];
}

// ---------------------------------------------------------------------------
// 5) edge scatter: agg[dst] += hw[src] * dinv[src]*dinv[dst]
//    32 lanes per edge, float4 per lane -> one coalesced 512B row per wave
// ---------------------------------------------------------------------------
__global__ void gcn_aggregate(const float* __restrict__ hw,
                              const int* __restrict__ src,
                              const int* __restrict__ dst,
                              const float* __restrict__ dinv,
                              float* __restrict__ agg, int n_edges) {
    int gid = blockIdx.x * blockDim.x + threadIdx.x;
    int e   = gid >> 5;
    if (e >= n_edges) return;
    int lane = gid & 31;
    int s = src[e];
    int d = dst[e];
    float nrm = dinv[s] * dinv[d];
    const float4 v = ((const float4*)(hw + (size_t)s * DIM))[lane];
    float* ap = agg + (size_t)d * DIM + lane * 4;
    unsafeAtomicAdd(ap + 0, v.x * nrm);
    unsafeAtomicAdd(ap + 1, v.y * nrm);
    unsafeAtomicAdd(ap + 2, v.z * nrm);
    unsafeAtomicAdd(ap + 3, v.w * nrm);
}

// ---------------------------------------------------------------------------
// 6) epilogue: add self-loop term + bias, BN(eval) + ReLU, zero agg for next
// ---------------------------------------------------------------------------
__global__ void bn_relu_epilogue(float4* __restrict__ agg,
                                 const float4* __restrict__ hw,
                                 const float* __restrict__ dinv,
                                 const float4* __restrict__ bias,
                                 const float4* __restrict__ gamma,
                                 const float4* __restrict__ beta,
                                 const float4* __restrict__ rmean,
                                 const float4* __restrict__ rvar,
                                 float4* __restrict__ out,
                                 int n_nodes, int zero_next) {
    int j = blockIdx.x * blockDim.x + threadIdx.x;   // float4 index
    if (j >= n_nodes * (DIM / 4)) return;
    int i  = j >> 5;          // node
    int dq = j & 31;          // float4 column within row

    float di = dinv[i];
    float sl = di * di;       // self-loop norm
    float4 a = agg[j], w = hw[j];
    float4 bb = bias[dq], gg = gamma[dq], be = beta[dq], mm = rmean[dq], vv = rvar[dq];

    float4 r;
    r.x = gg.x * ((a.x + w.x * sl + bb.x) - mm.x) * rsqrtf(vv.x + BN_EPS) + be.x;
    r.y = gg.y * ((a.y + w.y * sl + bb.y) - mm.y) * rsqrtf(vv.y + BN_EPS) + be.y;
    r.z = gg.z * ((a.z + w.z * sl + bb.z) - mm.z) * rsqrtf(vv.z + BN_EPS) + be.z;
    r.w = gg.w * ((a.w + w.w * sl + bb.w) - mm.w) * rsqrtf(vv.w + BN_EPS) + be.w;
    r.x = fmaxf(r.x, 0.f); r.y = fmaxf(r.y, 0.f);
    r.z = fmaxf(r.z, 0.f); r.w = fmaxf(r.w, 0.f);
    out[j] = r;
    if (zero_next) agg[j] = make_float4(0.f, 0.f, 0.f, 0.f);
}

// ---------------------------------------------------------------------------
extern "C" void kernel_launch(void* const* d_in, const int* in_sizes, int n_in,
                              void* d_out, int out_size, void* d_ws, size_t ws_size,
                              hipStream_t stream) {
    const float* x     = (const float*)d_in[0];
    const int*   ei    = (const int*)d_in[1];   // [2, E] row-major
    /* batch d_in[2] unused by reference forward */
    const float* Wall  = (const float*)d_in[3]; // [3,128,128]
    const float* ball  = (const float*)d_in[4]; // [3,128]
    const float* gall  = (const float*)d_in[5];
    const float* btall = (const float*)d_in[6];
    const float* rmall = (const float*)d_in[7];
    const float* rvall = (const float*)d_in[8];
    float*       outp  = (float*)d_out;

    const int n_nodes = in_sizes[0] / DIM;      // 100000
    const int n_edges = in_sizes[1] / 2;        // 1600000
    const int* src = ei;
    const int* dst = ei + n_edges;

    // workspace layout (floats)
    float* ws = (float*)d_ws;
    size_t off = 0;
    float* dinv = ws + off; off += ((size_t)n_nodes + 511) & ~(size_t)511;
    float* hw   = ws + off; off += (size_t)n_nodes * DIM;
    float* agg  = ws + off; off += (size_t)n_nodes * DIM;
    float* hbuf = ws + off; off += (size_t)n_nodes * DIM;
    (void)ws_size;

    const int nd = n_nodes * DIM;

    // degree + norm (computed once, reused for all 3 layers)
    init_deg_agg<<<(nd + 255) / 256, 256, 0, stream>>>(dinv, agg, n_nodes);
    accum_deg<<<(n_edges + 255) / 256, 256, 0, stream>>>(dst, dinv, n_edges);
    finish_dinv<<<(n_nodes + 255) / 256, 256, 0, stream>>>(dinv, n_nodes);

    const int gemm_blocks = n_nodes / 16;             // 6250 (exact)
    const int agg_blocks  = (n_edges * 32 + 255) / 256;
    const int epi_blocks  = (n_nodes * (DIM / 4) + 255) / 256;

    const float* h_in = x;
    for (int l = 0; l < 3; ++l) {
        const float* Wl = Wall + (size_t)l * DIM * DIM;
        float* h_out = (l == 2) ? outp : hbuf;

        gemm_wmma_f32<<<gemm_blocks, 256, 0, stream>>>(h_in, Wl, hw, n_nodes);
        gcn_aggregate<<<agg_blocks, 256, 0, stream>>>(hw, src, dst, dinv, agg, n_edges);
        bn_relu_epilogue<<<epi_blocks, 256, 0, stream>>>(
            (float4*)agg, (const float4*)hw, dinv,
            (const float4*)(ball  + (size_t)l * DIM),
            (const float4*)(gall  + (size_t)l * DIM),
            (const float4*)(btall + (size_t)l * DIM),
            (const float4*)(rmall + (size_t)l * DIM),
            (const float4*)(rvall + (size_t)l * DIM),
            (float4*)h_out, n_nodes, (l < 2) ? 1 : 0);

        h_in = hbuf;
    }
}